// FastContrastPixelCorrectCBL_21500606284461
// MI455X (gfx1250) — compile-verified
//
#include <hip/hip_runtime.h>
#include <math.h>

typedef float v2f __attribute__((ext_vector_type(2)));
typedef float v8f __attribute__((ext_vector_type(8)));

#define Bc 2
#define Cc 256
#define Hc 96
#define Wc 96
#define TILE 16
#define PR 5        // patch rows (dy = -2..2)
#define PCOL 20     // patch cols (dx = -2..17 relative to w0)
#define CHUNK 128   // channels per LDS pass
#define CPAD 132    // padded channel stride (132 % 64 == 4 -> conflict-free frags)
#define NPIX (Bc*Hc*Wc)

// ---------------- Kernel 1: per-pixel norm^2, argmax pred, packed code ----
__global__ __launch_bounds__(256)
void prep_kernel(const float* __restrict__ er, const float* __restrict__ logit,
                 const int* __restrict__ label, const int* __restrict__ gtb,
                 float* __restrict__ normsq, int* __restrict__ code,
                 float* __restrict__ acc)
{
    int p = blockIdx.x * 256 + threadIdx.x;
    if (p == 0) { acc[0] = 0.f; acc[1] = 0.f; acc[2] = 0.f; acc[3] = 0.f; }
    if (p >= NPIX) return;
    int b  = p / (Hc * Wc);
    int hw = p - b * (Hc * Wc);

    const float* base = er + (size_t)b * Cc * Hc * Wc + hw;
    float s = 0.f;
    #pragma unroll 8
    for (int ch = 0; ch < Cc; ++ch) {
        float v = base[(size_t)ch * (Hc * Wc)];
        s = fmaf(v, v, s);
    }
    normsq[p] = s;

    float l0 = logit[(size_t)(b * 2 + 0) * (Hc * Wc) + hw];
    float l1 = logit[(size_t)(b * 2 + 1) * (Hc * Wc) + hw];
    int pred = (l1 > l0) ? 1 : 0;               // argmax, ties -> 0
    int lab  = label[p] & 1;
    int g    = gtb[p];
    int edge = (g != 0 && g != 255) ? 1 : 0;
    code[p]  = lab | (pred << 1) | (edge << 2);
}

// ---------------- Kernel 2: main contrastive loss, WMMA banded GEMM ------
__global__ __launch_bounds__(160)
void loss_kernel(const float* __restrict__ er,
                 const float* __restrict__ normsq, const int* __restrict__ code,
                 float* __restrict__ acc)
{
    __shared__ float    sF[PR * PCOL * CPAD];   // feature patch, one channel chunk
    __shared__ float    sN[PR * PCOL];          // norm^2 patch
    __shared__ int      sC[PR * PCOL];          // packed label/pred/edge patch
    __shared__ float    sDots[TILE][26];        // 25 dots per pixel
    __shared__ float    sPosq[TILE][8];         // partial ||positive_sum||^2
    __shared__ unsigned sWnow[TILE];            // 25-bit same-class-correct masks
    __shared__ float    sLoss[TILE];
    __shared__ int      sCls[TILE];             // class id if masked-in, else -1

    const int tid  = threadIdx.x;
    const int lane = tid & 31;
    const int wv   = tid >> 5;        // wave id 0..4 == dy row of patch
    const int lo   = lane & 15;
    const int hiq  = lane >> 4;
    const int w0   = blockIdx.x * TILE;
    const int h    = blockIdx.y;
    const int b    = blockIdx.z;

    // ---- stage 0: code + norm patches (zero padded) ----
    for (int idx = tid; idx < PR * PCOL; idx += 160) {
        int r = idx / PCOL, cl = idx - r * PCOL;
        int hr = h - 2 + r, wn = w0 - 2 + cl;
        bool ok = (hr >= 0) & (hr < Hc) & (wn >= 0) & (wn < Wc);
        int p = (b * Hc + hr) * Wc + wn;
        sN[idx] = ok ? normsq[p] : 0.f;
        sC[idx] = ok ? code[p]   : 0;
    }
    __syncthreads();

    if (tid < TILE) {
        int L = sC[2 * PCOL + tid + 2] & 1;
        unsigned m = 0;
        #pragma unroll
        for (int k = 0; k < 25; ++k) {
            int cn = sC[(k / 5) * PCOL + tid + (k % 5)];
            if ((cn & 1) == L && ((cn >> 1) & 1) == L) m |= (1u << k);
        }
        sWnow[tid] = m;
    }
    __syncthreads();

    v8f d0 = {0.f,0.f,0.f,0.f,0.f,0.f,0.f,0.f};
    v8f d1 = {0.f,0.f,0.f,0.f,0.f,0.f,0.f,0.f};
    float posq = 0.f;

    for (int ch0 = 0; ch0 < Cc; ch0 += CHUNK) {
        // ---- load feature chunk into LDS (zero padded) ----
        for (int idx = tid; idx < CHUNK * PR * PCOL; idx += 160) {
            int ch  = idx / (PR * PCOL);
            int rem = idx - ch * (PR * PCOL);
            int r = rem / PCOL, cl = rem - r * PCOL;
            int hr = h - 2 + r, wn = w0 - 2 + cl;
            float v = 0.f;
            if (hr >= 0 && hr < Hc && wn >= 0 && wn < Wc)
                v = er[((size_t)(b * Cc + ch0 + ch) * Hc + hr) * Wc + wn];
            sF[rem * CPAD + ch] = v;
        }
        __syncthreads();

        // ---- WMMA banded GEMM: A = 16 anchors, B = 20 shifted cols (2 tiles)
        // fp32 A 16x4 frag: lane -> M, lane-half -> {K0,K1}/{K2,K3}
        // fp32 B 4x16 frag: lane -> N, lane-half -> K rows analogously
        const float* Ab  = &sF[(2 * PCOL + lo + 2) * CPAD];
        const float* B0b = &sF[(wv * PCOL + lo    ) * CPAD];
        const float* B1b = &sF[(wv * PCOL + lo + 4) * CPAD];
        #pragma unroll 8
        for (int kk = 0; kk < CHUNK; kk += 4) {
            int co = kk + 2 * hiq;
            v2f a  = *(const v2f*)(Ab  + co);
            v2f b0 = *(const v2f*)(B0b + co);
            v2f b1 = *(const v2f*)(B1b + co);
            d0 = __builtin_amdgcn_wmma_f32_16x16x4_f32(false, a, false, b0,
                                                       (short)0, d0, false, false);
            d1 = __builtin_amdgcn_wmma_f32_16x16x4_f32(false, a, false, b1,
                                                       (short)0, d1, false, false);
        }

        // ---- positive-sum vector partial norms (threads 0..127) ----
        if (tid < 128) {
            int i = tid >> 3, s = tid & 7;
            unsigned m = sWnow[i];
            for (int c = s * 16; c < s * 16 + 16; ++c) {
                float pv = 0.f;
                #pragma unroll
                for (int k = 0; k < 25; ++k) {
                    if (k == 12) continue;            // center excluded
                    if (m & (1u << k))
                        pv += sF[((k / 5) * PCOL + i + (k % 5)) * CPAD + c];
                }
                posq = fmaf(pv, pv, posq);
            }
        }
        __syncthreads();   // before next chunk overwrites sF
    }

    // ---- scatter band of D tiles into dots[pixel][25] ----
    {
        int dy5 = wv * 5;
        #pragma unroll
        for (int j = 0; j < 8; ++j) {
            int M = j + 8 * hiq;                  // pixel index
            int N = lo;                           // patch column (tile-local)
            int dxi0 = N - M;                     // tile0: cols 0..15
            if (dxi0 >= 0 && dxi0 <= 4) sDots[M][dy5 + dxi0] = d0[j];
            int dxi1 = N + 4 - M;                 // tile1: cols 4..19 (take 16..19)
            if (N >= 12 && dxi1 >= 0 && dxi1 <= 4) sDots[M][dy5 + dxi1] = d1[j];
        }
    }
    if (tid < 128) sPosq[tid >> 3][tid & 7] = posq;
    __syncthreads();

    // ---- per-pixel 26-way InfoNCE loss ----
    if (tid < TILE) {
        int i  = tid;
        int cc = sC[2 * PCOL + i + 2];
        int L = cc & 1, P = (cc >> 1) & 1, edge = (cc >> 2) & 1;
        float ind = (P == L) ? 1.f : 0.f;
        unsigned wn = sWnow[i];
        float a_norm = ind * sqrtf(sN[2 * PCOL + i + 2]);
        float sum_pd = 0.f;
        float lg[25];
        int count = 0;
        #pragma unroll
        for (int k = 0; k < 25; ++k) {
            int off = (k / 5) * PCOL + i + (k % 5);
            int cn  = sC[off];
            int lab = cn & 1, pr = (cn >> 1) & 1;
            if (k != 12 && lab == L) count++;
            float dfin = ind * sDots[i][k];
            if (k != 12 && (wn & (1u << k))) sum_pd += dfin;
            float wneg = (lab == (1 - L) && pr == (1 - L)) ? 2.f : 0.f;
            float nk   = sqrtf(sN[off]);
            lg[k] = ((dfin * wneg) / (a_norm * nk * wneg + 1e-8f)) * 10.f; // /TEMP
        }
        float pq = 0.f;
        #pragma unroll
        for (int s = 0; s < 8; ++s) pq += sPosq[i][s];
        float p_norm  = sqrtf(pq) * (1.f / 25.f);
        float apos    = sum_pd * (1.f / 25.f);
        float pos_cos = apos / (a_norm * p_norm + 1e-8f);
        float l0 = pos_cos * 10.f;
        float mx = l0;
        #pragma unroll
        for (int k = 0; k < 25; ++k) mx = fmaxf(mx, lg[k]);
        float sme = __expf(l0 - mx);
        #pragma unroll
        for (int k = 0; k < 25; ++k) sme += __expf(lg[k] - mx);
        sLoss[i] = mx + logf(sme) - l0;
        sCls[i]  = ((count >= 1) && edge) ? L : -1;
    }
    __syncthreads();

    if (tid == 0) {
        float s0 = 0.f, s1 = 0.f; int c0 = 0, c1 = 0;
        #pragma unroll
        for (int i = 0; i < TILE; ++i) {
            if (sCls[i] == 0)      { s0 += sLoss[i]; c0++; }
            else if (sCls[i] == 1) { s1 += sLoss[i]; c1++; }
        }
        if (c0) { atomicAdd(&acc[0], s0); atomicAdd(&acc[1], (float)c0); }
        if (c1) { atomicAdd(&acc[2], s1); atomicAdd(&acc[3], (float)c1); }
    }
}

// ---------------- Kernel 3: finalize scalar ------------------------------
__global__ void finalize_kernel(const float* __restrict__ acc, float* __restrict__ out)
{
    out[0] = acc[0] / fmaxf(acc[1], 1.f) + acc[2] / fmaxf(acc[3], 1.f);
}

extern "C" void kernel_launch(void* const* d_in, const int* in_sizes, int n_in,
                              void* d_out, int out_size, void* d_ws, size_t ws_size,
                              hipStream_t stream)
{
    const float* er    = (const float*)d_in[0];
    const float* logit = (const float*)d_in[1];
    const int*   label = (const int*)d_in[2];
    const int*   gtb   = (const int*)d_in[3];

    float* normsq = (float*)d_ws;                               // NPIX floats
    int*   code   = (int*)((char*)d_ws + (size_t)NPIX * 4);     // NPIX ints
    float* acc    = (float*)((char*)d_ws + (size_t)NPIX * 8);   // 4 floats

    prep_kernel<<<(NPIX + 255) / 256, 256, 0, stream>>>(er, logit, label, gtb,
                                                        normsq, code, acc);
    dim3 grid(Wc / TILE, Hc, Bc);   // 6 x 96 x 2
    loss_kernel<<<grid, 160, 0, stream>>>(er, normsq, code, acc);
    finalize_kernel<<<1, 1, 0, stream>>>(acc, (float*)d_out);
}